// TropicalREWAHead_84293028151339
// MI455X (gfx1250) — compile-verified
//
#include <hip/hip_runtime.h>

#define D_MODEL 1024
#define D_HEAD  64
#define M_BITS  32
#define SEQ_N   2048
#define SEQ_M   2048
#define NBATCH  2

typedef float v2f __attribute__((ext_vector_type(2)));
typedef float v8f __attribute__((ext_vector_type(8)));

// ---------------------------------------------------------------------------
// CDNA5 async global->LDS copy (per-lane global AND LDS addresses, so it can
// transpose on the fly). Tracked by ASYNCcnt.
// ---------------------------------------------------------------------------
__device__ __forceinline__ void async_cp_b32(unsigned lds_off, const float* g) {
  asm volatile("global_load_async_to_lds_b32 %0, %1, off"
               :: "v"(lds_off), "v"(g) : "memory");
}
#if __has_builtin(__builtin_amdgcn_s_wait_asynccnt)
#define WAIT_ASYNC(n) __builtin_amdgcn_s_wait_asynccnt(n)
#else
#define WAIT_ASYNC(n) asm volatile("s_wait_asynccnt %0" :: "i"(n) : "memory")
#endif

// ---------------------------------------------------------------------------
// Projection GEMM: Y[rows, cols] = X[rows, 1024] @ W[1024, cols], cols in {32,64}
// ---------------------------------------------------------------------------
#define PROJ_TR 32
#define PROJ_KC 64

__global__ __launch_bounds__(256)
void proj_kernel(const float* __restrict__ X, const float* __restrict__ W,
                 float* __restrict__ Y, int cols) {
  __shared__ float Xs[PROJ_TR][PROJ_KC];
  __shared__ float Ws[PROJ_KC][64];
  const int tid  = threadIdx.x;
  const int row0 = blockIdx.x * PROJ_TR;
  const int j     = tid % cols;
  const int rstep = 256 / cols;
  const int rb    = tid / cols;
  const int nout  = (PROJ_TR * cols) >> 8;

  float acc[8];
  #pragma unroll
  for (int i = 0; i < 8; ++i) acc[i] = 0.f;

  for (int k0 = 0; k0 < D_MODEL; k0 += PROJ_KC) {
    for (int e = tid; e < PROJ_TR * PROJ_KC; e += 256) {
      int r = e >> 6, k = e & 63;
      Xs[r][k] = X[(size_t)(row0 + r) * D_MODEL + k0 + k];
    }
    for (int e = tid; e < PROJ_KC * cols; e += 256) {
      int kk = e / cols, jj = e % cols;
      Ws[kk][jj] = W[(size_t)(k0 + kk) * cols + jj];
    }
    __syncthreads();
    for (int i = 0; i < nout; ++i) {
      int r = rb + rstep * i;
      float a = 0.f;
      #pragma unroll 8
      for (int kk = 0; kk < PROJ_KC; ++kk)
        a = fmaf(Xs[r][kk], Ws[kk][j], a);
      acc[i] += a;
    }
    __syncthreads();
  }
  for (int i = 0; i < nout; ++i) {
    int r = rb + rstep * i;
    Y[(size_t)(row0 + r) * cols + j] = acc[i];
  }
}

// ---------------------------------------------------------------------------
// Main fused kernel. One block (256 threads, 8 waves) owns 16 query rows.
//   pass1: async double-buffered Kt staging + min-plus scores into LDS tile
//   pass2: softmax in LDS, single attn write to HBM
//   pass3: Out = P @ Vp via V_WMMA_F32_16X16X4_F32
// ---------------------------------------------------------------------------
#define TQ   16
#define MC   64          // Kt chunk (columns) per pipeline stage
#define NCH  (SEQ_M / MC)
#define SST  2052        // stile row stride (2052 % 64 == 4; mult of 4 for b128)
#define KTS  68          // transposed-Kt row stride (68 % 64 == 4; mult of 4)

struct MainSmem {
  float stile[TQ * SST];                 // 131,328 B: scores then probs
  union {
    float ktile[2][M_BITS * KTS];        // 17,408 B: double-buffered Kt^T
    float pbuf[4 * 32 * 8];              // 4,096 B: WMMA partial-C exchange
  } u;
  float rinv[TQ];
};                                       // total 148,800 B (< 320 KB/WGP; 2 WGs fit)

__global__ __launch_bounds__(256)
void trewa_main(const float* __restrict__ Qt, const float* __restrict__ Kt,
                const float* __restrict__ Vp, const float* __restrict__ temp,
                float* __restrict__ Out, float* __restrict__ Attn) {
  __shared__ MainSmem sm;
  const int tid  = threadIdx.x;
  const int lane = tid & 31;
  const int wave = tid >> 5;
  const int tilesPerB = SEQ_N / TQ;
  const int b  = blockIdx.x / tilesPerB;
  const int n0 = (blockIdx.x % tilesPerB) * TQ;

  const float tv      = temp[0];
  const float inv_tau = 1.0f / log1pf(expf(tv));

  const int r = tid >> 4;     // query row owned by this thread
  const int c = tid & 15;     // column lane within the row

  // Qt row in registers
  float q[M_BITS];
  {
    const float4* qv = (const float4*)(Qt + (size_t)(b * SEQ_N + n0 + r) * M_BITS);
    #pragma unroll
    for (int t = 0; t < 8; ++t) {
      float4 v = qv[t];
      q[4*t+0] = v.x; q[4*t+1] = v.y; q[4*t+2] = v.z; q[4*t+3] = v.w;
    }
  }

  const float* ktB = Kt + (size_t)b * SEQ_M * M_BITS;
  float rmax = -__builtin_inff();

  // Issue async transpose-copy of one 64-row Kt chunk into ktile[buf].
  // Thread copies Kt[m0 + wave + 8t][lane] -> ktile[buf][lane*KTS + wave + 8t]
  // (global side: 32 lanes read consecutive floats -> coalesced 128B).
  auto issue_chunk = [&](int m0, int buf) {
    #pragma unroll
    for (int t = 0; t < 8; ++t) {
      int ml = wave + 8 * t;
      unsigned loff = (unsigned)(size_t)&sm.u.ktile[buf][lane * KTS + ml];
      async_cp_b32(loff, ktB + (size_t)(m0 + ml) * M_BITS + lane);
    }
  };

  // ---- pass 1: software-pipelined tropical min-plus ----
  issue_chunk(0, 0);
  for (int ci = 0; ci < NCH; ++ci) {
    const int cur = ci & 1;
    if (ci + 1 < NCH) {
      issue_chunk((ci + 1) * MC, cur ^ 1);
      WAIT_ASYNC(8);            // in-order completion => current chunk landed
    } else {
      WAIT_ASYNC(0);
    }
    __syncthreads();            // all waves' chunk data visible

    const float* kt = sm.u.ktile[cur];
    const int m0 = ci * MC;
    const int mbase = 4 * c;    // 4 consecutive columns per thread
    float mn0 = __builtin_inff(), mn1 = mn0, mn2 = mn0, mn3 = mn0;
    #pragma unroll
    for (int k = 0; k < M_BITS; ++k) {
      float4 kv = *(const float4*)&kt[k * KTS + mbase];   // conflict-free b128
      mn0 = fminf(mn0, q[k] + kv.x);
      mn1 = fminf(mn1, q[k] + kv.y);
      mn2 = fminf(mn2, q[k] + kv.z);
      mn3 = fminf(mn3, q[k] + kv.w);
    }
    float4 s4 = make_float4(-mn0, -mn1, -mn2, -mn3);
    rmax = fmaxf(rmax, fmaxf(fmaxf(s4.x, s4.y), fmaxf(s4.z, s4.w)));
    *(float4*)&sm.stile[r * SST + m0 + mbase] = s4;
    __syncthreads();            // everyone done reading ktile[cur] before reuse
  }

  // row-max across the 16 lanes of each row
  for (int off = 8; off > 0; off >>= 1)
    rmax = fmaxf(rmax, __shfl_xor(rmax, off, 16));

  // ---- pass 2: exp + rowsum (thread revisits only columns it wrote? no --
  // coverage is per-row either way; stride-16 keeps the attn write coalesced)
  float psum = 0.f;
  for (int jcol = 0; jcol < SEQ_M / 16; ++jcol) {
    int m = jcol * 16 + c;
    float s = sm.stile[r * SST + m];
    float p = __expf((s - rmax) * inv_tau);
    sm.stile[r * SST + m] = p;            // keep UNnormalized; scale out later
    psum += p;
  }
  for (int off = 8; off > 0; off >>= 1)
    psum += __shfl_xor(psum, off, 16);
  float invs = 1.0f / psum;
  if (c == 0) sm.rinv[r] = invs;

  {
    float* arow = Attn + (size_t)b * SEQ_N * SEQ_M + (size_t)(n0 + r) * SEQ_M;
    for (int jcol = 0; jcol < SEQ_M / 16; ++jcol) {
      int m = jcol * 16 + c;
      arow[m] = sm.stile[r * SST + m] * invs;
    }
  }
  __syncthreads();   // probs + rinv visible to all waves

  // ---- pass 3: Out[16,64] = P[16,2048] @ Vp[2048,64] via WMMA f32 16x16x4 ----
  {
    const int hb   = wave & 3;
    const int kh   = wave >> 2;
    const int h0   = hb * 16;
    const int half = lane >> 4;
    const int mrow = lane & 15;
    const float* vpB = Vp + (size_t)b * SEQ_M * D_HEAD;

    v8f cacc = {0.f, 0.f, 0.f, 0.f, 0.f, 0.f, 0.f, 0.f};
    const int kbeg = kh * (SEQ_M / 2);
    const int kend = kbeg + SEQ_M / 2;
    for (int k0 = kbeg; k0 < kend; k0 += 4) {
      const float2 av = *(const float2*)&sm.stile[mrow * SST + k0 + 2 * half];
      v2f a;  a.x = av.x;  a.y = av.y;
      v2f bf;
      bf.x = vpB[(size_t)(k0 + 2 * half)     * D_HEAD + h0 + mrow];
      bf.y = vpB[(size_t)(k0 + 2 * half + 1) * D_HEAD + h0 + mrow];
      cacc = __builtin_amdgcn_wmma_f32_16x16x4_f32(false, a, false, bf,
                                                   (short)0, cacc, false, false);
    }

    if (kh == 1) {
      #pragma unroll
      for (int i = 0; i < 8; ++i)
        sm.u.pbuf[(hb * 32 + lane) * 8 + i] = cacc[i];
    }
    __syncthreads();
    if (kh == 0) {
      #pragma unroll
      for (int i = 0; i < 8; ++i)
        cacc[i] += sm.u.pbuf[(hb * 32 + lane) * 8 + i];
      float* outB = Out + (size_t)b * SEQ_N * D_HEAD;
      #pragma unroll
      for (int i = 0; i < 8; ++i) {
        int row = i + 8 * half;
        float val = cacc[i] * sm.rinv[row];
        outB[(size_t)(n0 + row) * D_HEAD + h0 + mrow] = val;
      }
    }
  }
}

// ---------------------------------------------------------------------------
extern "C" void kernel_launch(void* const* d_in, const int* in_sizes, int n_in,
                              void* d_out, int out_size, void* d_ws, size_t ws_size,
                              hipStream_t stream) {
  const float* Q    = (const float*)d_in[0];
  const float* K    = (const float*)d_in[1];
  const float* V    = (const float*)d_in[2];
  const float* Wq   = (const float*)d_in[3];
  const float* Wk   = (const float*)d_in[4];
  const float* Wv   = (const float*)d_in[5];
  const float* temp = (const float*)d_in[6];

  float* out  = (float*)d_out;                               // [2,2048,64]
  float* attn = out + (size_t)NBATCH * SEQ_N * D_HEAD;       // [2,2048,2048]

  float* ws = (float*)d_ws;
  float* qt = ws;
  float* kt = qt + (size_t)NBATCH * SEQ_N * M_BITS;
  float* vp = kt + (size_t)NBATCH * SEQ_M * M_BITS;

  dim3 blk(256);
  proj_kernel<<<dim3((NBATCH * SEQ_N) / PROJ_TR), blk, 0, stream>>>(Q, Wq, qt, M_BITS);
  proj_kernel<<<dim3((NBATCH * SEQ_M) / PROJ_TR), blk, 0, stream>>>(K, Wk, kt, M_BITS);
  proj_kernel<<<dim3((NBATCH * SEQ_M) / PROJ_TR), blk, 0, stream>>>(V, Wv, vp, D_HEAD);
  trewa_main<<<dim3(NBATCH * (SEQ_N / TQ)), blk, 0, stream>>>(qt, kt, vp, temp, out, attn);
}